// StructurePredictor_86870008529456
// MI455X (gfx1250) — compile-verified
//
#include <hip/hip_runtime.h>
#include <hip/hip_bf16.h>
#include <math.h>

typedef __attribute__((ext_vector_type(2))) float f32x2;
typedef __attribute__((ext_vector_type(8))) float f32x8;

#define NSEQ 2048

// ---------------------------------------------------------------------------
// fp32 WMMA GEMM, 16x64 output strip per wave (4 accumulators reuse one A frag).
// C[M x N] = relu?( scale * (A[M x K] @ B[K x N]) + bias )
// TRANSB: B[k][n] accessed as Bm[n*ldB + k] (contiguous in k -> b64 loads).
// Out-of-range columns: load addresses clamped (branch-free, EXEC stays all-1s
// for WMMA); stores are per-lane guarded. K must be a multiple of 4.
// All buffers < 2^32 bytes -> 32-bit element offsets (saddr + voffset form).
// Grid: x covers ceil(N/64) strips / blockDim.y waves, y = M/16. Block (32,W).
// ---------------------------------------------------------------------------
template <bool TRANSB>
__global__ void wmma_gemm_strip(const float* __restrict__ A, int ldA,
                                const float* __restrict__ Bm, int ldB,
                                const float* __restrict__ bias, float scale,
                                int relu, float* __restrict__ C, int ldC,
                                int N, int nValid, int K)
{
    int strip = blockIdx.x * blockDim.y + threadIdx.y;
    int col0 = strip * 64;
    if (col0 >= N) return;                    // wave-uniform exit
    int tileM = blockIdx.y;
    int lane = threadIdx.x;
    int half = lane >> 4;                     // 0 -> k,k+1   1 -> k+2,k+3
    int l16  = lane & 15;
    int row  = tileM * 16 + l16;

    bool cok[4];
    int  colv[4];
#pragma unroll
    for (int s = 0; s < 4; ++s) {
        int c = col0 + s * 16 + l16;
        cok[s]  = (c < nValid);
        colv[s] = cok[s] ? c : (nValid - 1);  // clamp: branch-free safe loads
    }

    // 32-bit element offsets from the base pointers
    unsigned aoff = (unsigned)(row * ldA + 2 * half);
    unsigned boff0[4], boff1[4];
#pragma unroll
    for (int s = 0; s < 4; ++s) {
        if (TRANSB) {
            boff0[s] = (unsigned)(colv[s] * ldB + 2 * half);
            boff1[s] = 0;                     // unused
        } else {
            boff0[s] = (unsigned)((2 * half) * ldB + colv[s]);
            boff1[s] = boff0[s] + (unsigned)ldB;
        }
    }
    const unsigned bstep = TRANSB ? 4u : (unsigned)(4 * ldB);

    f32x8 acc[4];
#pragma unroll
    for (int s = 0; s < 4; ++s)
        acc[s] = (f32x8){0.f, 0.f, 0.f, 0.f, 0.f, 0.f, 0.f, 0.f};

    for (int k = 0; k < K; k += 4) {
        f32x2 a;
        a.x = A[aoff];
        a.y = A[aoff + 1];
        aoff += 4;
#pragma unroll
        for (int s = 0; s < 4; ++s) {
            f32x2 b;
            if (TRANSB) {
                b.x = Bm[boff0[s]];
                b.y = Bm[boff0[s] + 1];
                boff0[s] += bstep;
            } else {
                b.x = Bm[boff0[s]];
                b.y = Bm[boff1[s]];
                boff0[s] += bstep;
                boff1[s] += bstep;
            }
            acc[s] = __builtin_amdgcn_wmma_f32_16x16x4_f32(
                         false, a, false, b, (short)0, acc[s], false, false);
        }
    }

#pragma unroll
    for (int s = 0; s < 4; ++s) {
        if (!cok[s]) continue;                // per-lane store guard only
        int c = col0 + s * 16 + l16;
        float bv = bias ? bias[c] : 0.f;
#pragma unroll
        for (int r = 0; r < 8; ++r) {
            int orow = tileM * 16 + r + 8 * half;   // D: vgpr r -> row r / r+8
            float val = acc[s][r] * scale + bv;
            if (relu) val = fmaxf(val, 0.f);
            C[(unsigned)(orow * ldC + c)] = val;
        }
    }
}

// ---------------------------------------------------------------------------
// Row-wise softmax over a [rows x N] matrix, in place. One block per row.
// ---------------------------------------------------------------------------
__global__ void softmax_rows_kernel(float* __restrict__ S, int N)
{
    float* r = S + (size_t)blockIdx.x * N;
    __shared__ float red[256];
    int t = threadIdx.x;

    float mx = -INFINITY;
    for (int j = t; j < N; j += 256) mx = fmaxf(mx, r[j]);
    red[t] = mx; __syncthreads();
    for (int s = 128; s > 0; s >>= 1) {
        if (t < s) red[t] = fmaxf(red[t], red[t + s]);
        __syncthreads();
    }
    mx = red[0];
    __syncthreads();

    float sum = 0.f;
    for (int j = t; j < N; j += 256) {
        float e = __expf(r[j] - mx);
        r[j] = e;
        sum += e;
    }
    red[t] = sum; __syncthreads();
    for (int s = 128; s > 0; s >>= 1) {
        if (t < s) red[t] += red[t + s];
        __syncthreads();
    }
    float inv = 1.f / red[0];
    for (int j = t; j < N; j += 256) r[j] *= inv;
}

// ---------------------------------------------------------------------------
// Tiny head: out[row, d] = sum_j X[row, j] * W[j, d] + bias[d], nOut <= 3.
// ---------------------------------------------------------------------------
__global__ void head_kernel(const float* __restrict__ X, int K,
                            const float* __restrict__ W, int nOut,
                            const float* __restrict__ bias,
                            float* __restrict__ out)
{
    int row = blockIdx.x;
    int t = threadIdx.x;
    __shared__ float red[3][256];
    float acc[3] = {0.f, 0.f, 0.f};
    for (int j = t; j < K; j += 256) {
        float xv = X[(size_t)row * K + j];
        for (int d = 0; d < nOut; ++d) acc[d] += xv * W[j * nOut + d];
    }
    for (int d = 0; d < 3; ++d) red[d][t] = acc[d];
    __syncthreads();
    for (int s = 128; s > 0; s >>= 1) {
        if (t < s)
            for (int d = 0; d < nOut; ++d) red[d][t] += red[d][t + s];
        __syncthreads();
    }
    if (t < nOut) out[(size_t)row * nOut + t] = red[t][0] + bias[t];
}

__global__ void zero_kernel(float* __restrict__ p, int n)
{
    int i = blockIdx.x * blockDim.x + threadIdx.x;
    if (i < n) p[i] = 0.f;
}

// ---------------------------------------------------------------------------
// One fused Adam refinement step. contact_map[q,k] = prob[q] (rank-1), so
//   g_i = (1/N^2) * sum_j (h'(p_i,d_ij) + h'(p_j,d_ij)) * (x_i - x_j)/d_ij
//   h'(p,d) = p*[d>8] - (1-p)*[d<12]
// plus the bond-length term. One block per point i; all of x staged in LDS.
// ---------------------------------------------------------------------------
__global__ void refine_step_kernel(const float* __restrict__ xin,
                                   float* __restrict__ xout,
                                   float* __restrict__ m, float* __restrict__ v,
                                   const float* __restrict__ p,
                                   float inv1mb1t, float inv1mb2t)
{
    const int N = NSEQ;
    __shared__ float sx[NSEQ * 3];
    __shared__ float sp[NSEQ];
    __shared__ float rg[3][256];
    int i = blockIdx.x, t = threadIdx.x;

    for (int j = t; j < N; j += 256) {
        sx[3 * j + 0] = xin[3 * j + 0];
        sx[3 * j + 1] = xin[3 * j + 1];
        sx[3 * j + 2] = xin[3 * j + 2];
        sp[j] = p[j];
    }
    __syncthreads();

    float xi0 = sx[3 * i], xi1 = sx[3 * i + 1], xi2 = sx[3 * i + 2];
    float pi = sp[i];
    float g0 = 0.f, g1 = 0.f, g2 = 0.f;
    for (int j = t; j < N; j += 256) {
        float d0 = xi0 - sx[3 * j], d1 = xi1 - sx[3 * j + 1], d2 = xi2 - sx[3 * j + 2];
        float dist = sqrtf(d0 * d0 + d1 * d1 + d2 * d2 + 1e-12f);
        float pj = sp[j];
        float hi = (dist > 8.f ? pi : 0.f) - (dist < 12.f ? (1.f - pi) : 0.f);
        float hj = (dist > 8.f ? pj : 0.f) - (dist < 12.f ? (1.f - pj) : 0.f);
        float w = (hi + hj) / dist;
        g0 += w * d0; g1 += w * d1; g2 += w * d2;
    }
    rg[0][t] = g0; rg[1][t] = g1; rg[2][t] = g2;
    __syncthreads();
    for (int s = 128; s > 0; s >>= 1) {
        if (t < s) {
            rg[0][t] += rg[0][t + s];
            rg[1][t] += rg[1][t + s];
            rg[2][t] += rg[2][t + s];
        }
        __syncthreads();
    }

    if (t == 0) {
        const float invN2 = 1.f / ((float)N * (float)N);
        float g[3] = { rg[0][0] * invN2, rg[1][0] * invN2, rg[2][0] * invN2 };
        const float bw = 0.1f / (float)(N - 1);
        if (i > 0) {
            float d0 = xi0 - sx[3 * (i - 1)], d1 = xi1 - sx[3 * (i - 1) + 1],
                  d2 = xi2 - sx[3 * (i - 1) + 2];
            float b = sqrtf(d0 * d0 + d1 * d1 + d2 * d2 + 1e-12f);
            if (b > 4.f) { float w = bw / b; g[0] += w * d0; g[1] += w * d1; g[2] += w * d2; }
        }
        if (i < N - 1) {
            float d0 = xi0 - sx[3 * (i + 1)], d1 = xi1 - sx[3 * (i + 1) + 1],
                  d2 = xi2 - sx[3 * (i + 1) + 2];
            float b = sqrtf(d0 * d0 + d1 * d1 + d2 * d2 + 1e-12f);
            if (b > 4.f) { float w = bw / b; g[0] += w * d0; g[1] += w * d1; g[2] += w * d2; }
        }
        float xi[3] = { xi0, xi1, xi2 };
        for (int c = 0; c < 3; ++c) {
            float gc = g[c];
            float mc = 0.9f   * m[3 * i + c] + 0.1f   * gc;
            float vc = 0.999f * v[3 * i + c] + 0.001f * gc * gc;
            m[3 * i + c] = mc;
            v[3 * i + c] = vc;
            float mh = mc * inv1mb1t;
            float vh = vc * inv1mb2t;
            xout[3 * i + c] = xi[c] - 0.01f * mh / (sqrtf(vh) + 1e-8f);
        }
    }
}

// ---------------------------------------------------------------------------
extern "C" void kernel_launch(void* const* d_in, const int* in_sizes, int n_in,
                              void* d_out, int out_size, void* d_ws, size_t ws_size,
                              hipStream_t stream)
{
    (void)in_sizes; (void)n_in; (void)out_size; (void)ws_size;
    const float* features = (const float*)d_in[0];
    const float* bb_w1 = (const float*)d_in[1];
    const float* bb_b1 = (const float*)d_in[2];
    const float* bb_w2 = (const float*)d_in[3];
    const float* bb_b2 = (const float*)d_in[4];
    // d_in[5..8] = side_chain network: dead code, never affects the output
    const float* wqkv  = (const float*)d_in[9];
    const float* bqkv  = (const float*)d_in[10];
    const float* wo    = (const float*)d_in[11];
    const float* bo    = (const float*)d_in[12];
    const float* cm_w1 = (const float*)d_in[13];
    const float* cm_b1 = (const float*)d_in[14];
    const float* cm_w2 = (const float*)d_in[15];
    const float* cm_b2 = (const float*)d_in[16];
    const float* rb_w1 = (const float*)d_in[17];
    const float* rb_b1 = (const float*)d_in[18];
    const float* rb_w2 = (const float*)d_in[19];
    const float* rb_b2 = (const float*)d_in[20];
    // d_in[21..24] = rs_* head: dead code

    float* ws = (float*)d_ws;
    // workspace layout (floats); total ~32.6 MB (fits comfortably in 192MB L2)
    float* h1       = ws;                    // 2048*320  (reused as ctx later)
    float* bbf      = ws + 655360;           // 2048*320
    float* qkv      = ws + 1310720;          // 2048*960  (reused as attn_out later)
    float* scores   = ws + 3276800;          // 2048*2048 (per-head, reused)
    float* cmh      = ws + 7471104;          // 2048*160
    float* rbh      = ws + 7798784;          // 2048*160
    float* prob     = ws + 8126464;          // 2048
    float* xA       = ws + 8128512;          // 2048*3
    float* xB       = ws + 8134656;          // 2048*3
    float* adam_m   = ws + 8140800;          // 2048*3
    float* adam_v   = ws + 8146944;          // 2048*3
    float* ctx      = h1;                    // alias (h1 dead after backbone GEMM2)
    float* attn_out = qkv;                   // alias (qkv dead after head loop)

    dim3 blk(32, 4);

    // backbone: h1 = relu(features @ bb_w1 + b1);  bbf = h1 @ bb_w2 + b2
    wmma_gemm_strip<false><<<dim3(2, 128), blk, 0, stream>>>(
        features, 320, bb_w1, 320, bb_b1, 1.f, 1, h1, 320, 320, 320, 320);
    wmma_gemm_strip<false><<<dim3(2, 128), blk, 0, stream>>>(
        h1, 320, bb_w2, 320, bb_b2, 1.f, 0, bbf, 320, 320, 320, 320);

    // qkv = features @ wqkv + bqkv   (2048 x 960)
    wmma_gemm_strip<false><<<dim3(4, 128), blk, 0, stream>>>(
        features, 320, wqkv, 960, bqkv, 1.f, 0, qkv, 960, 960, 960, 320);

    // attention, one head at a time (scores buffer reused)
    const float scl = 1.0f / sqrtf(40.0f);
    for (int h = 0; h < 8; ++h) {
        // S = scale * Q_h @ K_h^T     (M=2048, N=2048, K=40; B transposed)
        wmma_gemm_strip<true><<<dim3(8, 128), blk, 0, stream>>>(
            qkv + h * 40, 960, qkv + 320 + h * 40, 960,
            nullptr, scl, 0, scores, 2048, 2048, 2048, 40);
        softmax_rows_kernel<<<2048, 256, 0, stream>>>(scores, 2048);
        // ctx_h = P @ V_h             (M=2048, N=40, K=2048)
        wmma_gemm_strip<false><<<dim3(1, 128), blk, 0, stream>>>(
            scores, 2048, qkv + 640 + h * 40, 960,
            nullptr, 1.f, 0, ctx + h * 40, 320, 40, 40, 2048);
    }

    // attn_out = ctx @ wo + bo
    wmma_gemm_strip<false><<<dim3(2, 128), blk, 0, stream>>>(
        ctx, 320, wo, 320, bo, 1.f, 0, attn_out, 320, 320, 320, 320);

    // contact mlp: cmh = relu(attn_out @ cm_w1 + b);  prob = cmh @ cm_w2 + b
    wmma_gemm_strip<false><<<dim3(1, 128), blk, 0, stream>>>(
        attn_out, 320, cm_w1, 160, cm_b1, 1.f, 1, cmh, 160, 160, 160, 320);
    head_kernel<<<2048, 256, 0, stream>>>(cmh, 160, cm_w2, 1, cm_b2, prob);

    // refiner backbone: rbh = relu(bbf @ rb_w1 + b);  x0 = rbh @ rb_w2 + b
    wmma_gemm_strip<false><<<dim3(1, 128), blk, 0, stream>>>(
        bbf, 320, rb_w1, 160, rb_b1, 1.f, 1, rbh, 160, 160, 160, 320);
    head_kernel<<<2048, 256, 0, stream>>>(rbh, 160, rb_w2, 3, rb_b2, xA);

    // Adam state init (ws is poisoned, and must be re-zeroed every call)
    zero_kernel<<<(12288 + 255) / 256, 256, 0, stream>>>(adam_m, 12288);

    // 100 sequential refinement steps, ping-pong; last step writes d_out
    for (int t = 1; t <= 100; ++t) {
        const float* xin = (t & 1) ? xA : xB;
        float* xout = (t == 100) ? (float*)d_out : ((t & 1) ? xB : xA);
        float i1 = 1.f / (1.f - powf(0.9f,   (float)t));
        float i2 = 1.f / (1.f - powf(0.999f, (float)t));
        refine_step_kernel<<<2048, 256, 0, stream>>>(
            xin, xout, adam_m, adam_v, prob, i1, i2);
    }
}